// MappedMaxUnpool_58463094833216
// MI455X (gfx1250) — compile-verified
//
#include <hip/hip_runtime.h>

// MappedMaxUnpool scatter-add for MI455X (gfx1250).
// Shapes (fixed by reference): x/idx_mask (16,64,128,128), tables (128,128,4,4),
// out (16,64,256,256). Harness passes integer inputs as int32.

#define OUT_HW   (256 * 256)
#define OH_      128
#define OW_      128
#define KS       4
#define PS       4
#define BC_TOTAL (16 * 64)          // 1024 (b,c) planes
#define NCHUNK   16                 // bc chunks per oh row
#define BC_PER_BLOCK (BC_TOTAL / NCHUNK)   // 64
#define THREADS  256
#define ROW_ELEMS (OW_ * KS * PS)   // 2048 elements per (oh) table row

typedef __attribute__((address_space(3))) char lds_char_t;

// LDS byte offset of a __shared__ object (addrspacecast generic->local, then ptrtoint).
__device__ __forceinline__ unsigned lds_addr_of(const void* p) {
    return (unsigned)(unsigned long long)(lds_char_t*)p;
}

// Async DMA of 16 bytes global -> LDS, tracked by ASYNCcnt (CDNA5).
__device__ __forceinline__ void async_copy_b128(unsigned lds_off, const void* gptr) {
    unsigned long long ga = (unsigned long long)gptr;
    asm volatile("global_load_async_to_lds_b128 %0, %1, off"
                 :: "v"(lds_off), "v"(ga)
                 : "memory");
}

__device__ __forceinline__ void wait_asynccnt0() {
    asm volatile("s_wait_asynccnt 0" ::: "memory");
}

__global__ void __launch_bounds__(THREADS)
unpool_zero_kernel(float4* __restrict__ out, int n4) {
    int i = blockIdx.x * blockDim.x + threadIdx.x;
    const int stride = gridDim.x * blockDim.x;
    const float4 z = make_float4(0.f, 0.f, 0.f, 0.f);
    for (; i < n4; i += stride) out[i] = z;
}

__global__ void __launch_bounds__(THREADS)
unpool_scatter_kernel(const float* __restrict__ x,
                      const int*   __restrict__ idx_mask,
                      const int*   __restrict__ smap,
                      const float* __restrict__ sw,
                      float*       __restrict__ out) {
    __shared__ __align__(16) int   s_idx[ROW_ELEMS];  // 8 KB: sample_map row
    __shared__ __align__(16) float s_w  [ROW_ELEMS];  // 8 KB: interp_weights row

    const int tid  = threadIdx.x;
    const int oh   = blockIdx.x;                 // 0..127
    const int bc0  = blockIdx.y * BC_PER_BLOCK;  // chunk of (b*C+c) planes
    const int row  = oh * ROW_ELEMS;             // table row base (elements)

    // ---- Stage this oh-row's (sample_map, weights) tables into LDS via async DMA.
    {
        const unsigned lb_i = lds_addr_of(s_idx);
        const unsigned lb_w = lds_addr_of(s_w);
#pragma unroll
        for (int j = 0; j < 2; ++j) {
            const int e = (j * THREADS + tid) * 4;        // 4 dwords per b128
            async_copy_b128(lb_i + e * 4, smap + row + e);
            async_copy_b128(lb_w + e * 4, sw   + row + e);
        }
    }
    wait_asynccnt0();
    __syncthreads();

    const int ow  = tid & (OW_ - 1);   // lanes contiguous in ow -> coalesced
    const int sub = tid >> 7;          // 0..1: two bc planes per iteration

    const int4*   s_idx4 = (const int4*)s_idx;
    const float4* s_w4   = (const float4*)s_w;

#pragma unroll 1
    for (int it = 0; it < BC_PER_BLOCK / 2; ++it) {
        const int bc   = bc0 + it * 2 + sub;
        const int gidx = (bc * OH_ + oh) * OW_ + ow;   // < 2^24, int-safe

        // Prefetch next iteration's cachelines (global_prefetch_b8).
        if (it + 1 < BC_PER_BLOCK / 2) {
            __builtin_prefetch(x        + gidx + 2 * OH_ * OW_, 0, 3);
            __builtin_prefetch(idx_mask + gidx + 2 * OH_ * OW_, 0, 3);
        }

        const float v = x[gidx];
        const int   k = idx_mask[gidx] & (KS - 1);

        const int4   si = s_idx4[ow * KS + k];   // ds_load_b128
        const float4 wv = s_w4  [ow * KS + k];   // ds_load_b128

        float* base = out + bc * OUT_HW;
        atomicAdd(base + si.x, wv.x * v);        // global_atomic_add_f32 (no return)
        atomicAdd(base + si.y, wv.y * v);
        atomicAdd(base + si.z, wv.z * v);
        atomicAdd(base + si.w, wv.w * v);
    }
}

extern "C" void kernel_launch(void* const* d_in, const int* in_sizes, int n_in,
                              void* d_out, int out_size, void* d_ws, size_t ws_size,
                              hipStream_t stream) {
    const float* x        = (const float*)d_in[0];
    const int*   idx_mask = (const int*)  d_in[1];
    const int*   smap     = (const int*)  d_in[2];
    const float* sw       = (const float*)d_in[3];
    float*       out      = (float*)d_out;

    // 1) Zero the 256 MB output with float4 stores.
    const int n4 = out_size / 4;
    unpool_zero_kernel<<<8192, THREADS, 0, stream>>>((float4*)out, n4);

    // 2) Scatter-add: one block per (oh, bc-chunk); LDS-staged tables.
    dim3 grid(OH_, NCHUNK);
    unpool_scatter_kernel<<<grid, THREADS, 0, stream>>>(x, idx_mask, smap, sw, out);
}